// MF_71579924955683
// MI455X (gfx1250) — compile-verified
//
#include <hip/hip_runtime.h>

// diag(U @ V^T) via V_WMMA_F32_16X16X4_F32, K accumulated 4 at a time.
// Memory-bound embedding-gather dot product: 16.8 MB gathered -> ~0.75us floor
// at 23.3 TB/s. WMMA operand layout makes each lane's feed exactly one
// global_load_b64 from its (gathered) row per K-step. Diagonal extraction is
// a branchless cndmask select chain + one exec-masked coalesced store.

typedef __attribute__((ext_vector_type(2))) float v2f;
typedef __attribute__((ext_vector_type(8))) float v8f;

#define EMBED_DIM   128
#define ITEM_OFFSET 1000000ll
#define WAVES_PER_BLOCK 8
#define ROWS_PER_WAVE   16
#define ROWS_PER_BLOCK  (WAVES_PER_BLOCK * ROWS_PER_WAVE)  // 128

__global__ __launch_bounds__(256) void embed_dot_wmma(
    const long long* __restrict__ x,     // [batch, 2] int64 indices
    const float*     __restrict__ table, // [1.5M, 128] f32
    float*           __restrict__ out,   // [batch] f32
    int batch)
{
    const int lane    = threadIdx.x & 31;
    const int wave    = threadIdx.x >> 5;
    const int rowBase = blockIdx.x * ROWS_PER_BLOCK + wave * ROWS_PER_WAVE;
    const int myRow   = rowBase + (lane & 15);

    // Per-lane gather targets. Lane L feeds matrix row/col M=N=L%16;
    // its K pair within each K=4 step is 2*(L/16) + {0,1}.
    const long long uidx = x[2 * myRow];
    const long long iidx = x[2 * myRow + 1] + ITEM_OFFSET;
    const int       kSub = (lane >> 4) << 1;

    const float* up = table + uidx * EMBED_DIM + kSub;
    const float* ip = table + iidx * EMBED_DIM + kSub;

    v8f c = {};

    // 32 WMMA steps cover K=128. Each step: two 8B loads per lane
    // (A element pair from user row, B element pair from item row).
#pragma unroll 8
    for (int k = 0; k < EMBED_DIM; k += 4) {
        v2f a = *(const v2f*)(up + k);
        v2f b = *(const v2f*)(ip + k);
        // 8 args: (neg_a, A, neg_b, B, c_mod, C, reuse_a, reuse_b)
        c = __builtin_amdgcn_wmma_f32_16x16x4_f32(
                /*neg_a=*/false, a, /*neg_b=*/false, b,
                /*c_mod=*/(short)0, c, /*reuse_a=*/false, /*reuse_b=*/false);
    }

    // Diagonal of 16x16 f32 D lives at (VGPR r, lane r) and (VGPR r, lane 24+r).
    // For both holder lanes, (lane & 7) == r, so a branchless select chain
    // places each lane's diagonal value in y.
    const int sel = lane & 7;
    float y = c[0];
#pragma unroll
    for (int r = 1; r < 8; ++r) {
        y = (sel == r) ? c[r] : y;
    }

    // Active lanes 0..7 write y[rowBase+0..7]; lanes 24..31 write y[rowBase+8..15].
    // One exec-masked store, contiguous 64B across active lanes.
    if (lane < 8 || lane >= 24) {
        out[rowBase + (lane & 15)] = y;
    }
}

extern "C" void kernel_launch(void* const* d_in, const int* in_sizes, int n_in,
                              void* d_out, int out_size, void* d_ws, size_t ws_size,
                              hipStream_t stream) {
    const long long* x     = (const long long*)d_in[0]; // int64 [batch,2]
    const float*     table = (const float*)d_in[1];     // f32 [1.5M,128]
    float*           out   = (float*)d_out;             // f32 [batch]

    const int batch  = in_sizes[0] / 2;                 // 16384
    const int blocks = batch / ROWS_PER_BLOCK;          // 128

    embed_dot_wmma<<<blocks, 256, 0, stream>>>(x, table, out, batch);
}